// UnfoldindAndAttention_69999376990389
// MI455X (gfx1250) — compile-verified
//
#include <hip/hip_runtime.h>
#include <hip/hip_bf16.h>

typedef __attribute__((ext_vector_type(2))) float v2f;
typedef __attribute__((ext_vector_type(8))) float v8f;

// Reference constants: LAM=1.0, ALP=1/(LAM+1)=0.5, PROP_STEP=5
#define LAM_C 1.0f
#define ALP_C 0.5f
#define PROP_STEPS 5
constexpr float C1 = 1.0f - ALP_C - ALP_C * LAM_C;  // coefficient on Y (== 0 here)
constexpr float C2 = ALP_C * LAM_C;                  // coefficient on norm_AX(Y)
constexpr float C3 = ALP_C;                          // coefficient on X

// Packed edge encoding: (src << 5) | row, row in [0,15] real, row==16 => dummy
// (dummy contributes A==0 so the WMMA accumulates nothing; src forced to 0 so
//  the B-side gather stays in bounds).
#define DUMMY_EDGE 16
#define BPAD 16   // bucket edge-count padding (=> tail-free 4x-unrolled WMMA loop)

// ---------------- preprocessing kernels ----------------

__global__ void deg_kernel(const int* __restrict__ dst, int* __restrict__ deg, int E) {
    int i = blockIdx.x * blockDim.x + threadIdx.x;
    if (i < E) atomicAdd(&deg[dst[i]], 1);
}

// Single-block scan over NB buckets (bucket b = nodes [16b, 16b+16)).
// Bucket sizes are padded to a multiple of BPAD so every bucket start is a
// multiple of 16 edges (8B-aligned int2 loads, tail-free unroll in prop_kernel).
// Writes exclusive padded offsets boff[0..NB], cursor[b]=boff[b], real size bsz[b].
__global__ void scan_kernel(const int* __restrict__ deg,
                            int* __restrict__ boff, int* __restrict__ cursor,
                            int* __restrict__ bsz, int NB, int N) {
    __shared__ int sdata[1024];
    const int tid   = threadIdx.x;
    const int items = (NB + 1023) >> 10;   // buckets per thread (<= 8 for NB <= 8192)
    const int base  = tid * items;

    int locp[8];   // padded sizes
    int locr[8];   // real sizes
    int sum = 0;
    for (int i = 0; i < items && i < 8; ++i) {
        int b = base + i;
        int c = 0;
        if (b < NB) {
            int n0 = b << 4;
            for (int k = 0; k < 16; ++k) {
                int n = n0 + k;
                if (n < N) c += deg[n];
            }
        }
        locr[i] = c;
        locp[i] = (c + (BPAD - 1)) & ~(BPAD - 1);
        sum += locp[i];
    }
    sdata[tid] = sum;
    __syncthreads();
    // Hillis-Steele inclusive scan across 1024 threads
    for (int off = 1; off < 1024; off <<= 1) {
        int v = 0;
        if (tid >= off) v = sdata[tid - off];
        __syncthreads();
        sdata[tid] += v;
        __syncthreads();
    }
    int excl = sdata[tid] - sum;
    for (int i = 0; i < items && i < 8; ++i) {
        int b = base + i;
        if (b < NB) {
            boff[b]   = excl;
            cursor[b] = excl;
            bsz[b]    = locr[i];
            excl += locp[i];
        }
    }
    if (tid == 0) boff[NB] = sdata[1023];
}

// Counting-sort scatter of packed edges into their dst bucket.
__global__ void fill_kernel(const int* __restrict__ src, const int* __restrict__ dst,
                            int* __restrict__ cursor, int* __restrict__ ebuf, int E) {
    int i = blockIdx.x * blockDim.x + threadIdx.x;
    if (i < E) {
        int s = src[i], d = dst[i];
        int b = d >> 4;
        int pos = atomicAdd(&cursor[b], 1);
        ebuf[pos] = (s << 5) | (d & 15);
    }
}

// Fill the (< BPAD) padding slots of each bucket with DUMMY_EDGE.
__global__ void pad_kernel(const int* __restrict__ boff, const int* __restrict__ bsz,
                           int* __restrict__ ebuf, int NB) {
    int i = blockIdx.x * blockDim.x + threadIdx.x;
    int b = i >> 4;          // bucket
    int k = i & 15;          // pad slot candidate 0..15
    if (b < NB) {
        int size = bsz[b];
        int pads = ((size + (BPAD - 1)) & ~(BPAD - 1)) - size;   // 0..15
        if (k < pads) ebuf[boff[b] + size + k] = DUMMY_EDGE;
    }
}

// dinv = deg>0 ? rsqrt(deg) : 0 ; Y0 = X ; Z0 = X * dinv
__global__ void init_kernel(const float* __restrict__ X, const int* __restrict__ deg,
                            float* __restrict__ dinv,
                            float* __restrict__ Y, float* __restrict__ Z, int N) {
    int n = blockIdx.x * blockDim.x + threadIdx.x;
    if (n >= N) return;
    int d = deg[n];
    float di = (d > 0) ? rsqrtf((float)d) : 0.0f;
    dinv[n] = di;
    const float4* X4 = (const float4*)X;
    float4* Y4 = (float4*)Y;
    float4* Z4 = (float4*)Z;
    #pragma unroll
    for (int j = 0; j < 4; ++j) {
        float4 x = X4[n * 4 + j];
        Y4[n * 4 + j] = x;
        float4 z;
        z.x = x.x * di; z.y = x.y * di; z.z = x.z * di; z.w = x.w * di;
        Z4[n * 4 + j] = z;
    }
}

// ---------------- propagation: one wave per bucket of 16 dst nodes ----------------
// Segment-sum via chained V_WMMA_F32_16X16X4_F32:
//   A (16x4) = one-hot dst-row selector, B (4x16) = gathered Z[src] rows,
//   C (16x16) accumulates agg for the bucket's 16 nodes. No atomics, no LDS.
// K layout (wave32): lanes 0-15 supply K=0 (vgpr0) / K=1 (vgpr1);
//                    lanes 16-31 supply K=2 / K=3.
// 4 independent accumulators break the WMMA->WMMA RAW chain; bucket padding to
// 16 edges makes the x4-unrolled loop tail-free.
__global__ __launch_bounds__(256) void prop_kernel(
    const int* __restrict__ boff, const int* __restrict__ ebuf,
    const float* __restrict__ Zin, const float* __restrict__ Yin,
    const float* __restrict__ X, const unsigned char* __restrict__ mask,
    const float* __restrict__ dinv,
    float* __restrict__ Yout, float* __restrict__ Zout,
    int NB, int N) {

    const int wave = threadIdx.x >> 5;
    const int lane = threadIdx.x & 31;
    const int b = blockIdx.x * (blockDim.x >> 5) + wave;
    if (b >= NB) return;

    const int half = lane >> 4;   // which K-half this lane supplies
    const int mn   = lane & 15;   // A: row m ; B: feature column n ; C: column n

    const int start = boff[b];        // multiple of 16
    const int end   = boff[b + 1];    // multiple of 16

    v8f acc0 = {0.f, 0.f, 0.f, 0.f, 0.f, 0.f, 0.f, 0.f};
    v8f acc1 = {0.f, 0.f, 0.f, 0.f, 0.f, 0.f, 0.f, 0.f};
    v8f acc2 = {0.f, 0.f, 0.f, 0.f, 0.f, 0.f, 0.f, 0.f};
    v8f acc3 = {0.f, 0.f, 0.f, 0.f, 0.f, 0.f, 0.f, 0.f};

    // This lane's two packed edges of group g live at g*4 + 2*half (+1):
    // an 8B-aligned int2, identical address across each half-wave (broadcast).
    const int2* __restrict__ ep = (const int2*)ebuf;

    for (int e = start; e < end; e += 16) {
        const int q = (e >> 1) + half;      // int2 index of this lane's pair
        int2 p0 = ep[q + 0];
        int2 p1 = ep[q + 2];
        int2 p2 = ep[q + 4];
        int2 p3 = ep[q + 6];
        __builtin_prefetch(&ebuf[e + 128 + lane], 0, 1);   // speculative, safe

        // decode: row = p & 31 (16 => dummy, matches no mn), src = p >> 5
        {
            float bx = Zin[(int)(((unsigned)p0.x) >> 5) * 16 + mn];
            float by = Zin[(int)(((unsigned)p0.y) >> 5) * 16 + mn];
            v2f A = {((p0.x & 31) == mn) ? 1.0f : 0.0f,
                     ((p0.y & 31) == mn) ? 1.0f : 0.0f};
            v2f B = {bx, by};
            acc0 = __builtin_amdgcn_wmma_f32_16x16x4_f32(false, A, false, B,
                                                         (short)0, acc0, false, false);
        }
        {
            float bx = Zin[(int)(((unsigned)p1.x) >> 5) * 16 + mn];
            float by = Zin[(int)(((unsigned)p1.y) >> 5) * 16 + mn];
            v2f A = {((p1.x & 31) == mn) ? 1.0f : 0.0f,
                     ((p1.y & 31) == mn) ? 1.0f : 0.0f};
            v2f B = {bx, by};
            acc1 = __builtin_amdgcn_wmma_f32_16x16x4_f32(false, A, false, B,
                                                         (short)0, acc1, false, false);
        }
        {
            float bx = Zin[(int)(((unsigned)p2.x) >> 5) * 16 + mn];
            float by = Zin[(int)(((unsigned)p2.y) >> 5) * 16 + mn];
            v2f A = {((p2.x & 31) == mn) ? 1.0f : 0.0f,
                     ((p2.y & 31) == mn) ? 1.0f : 0.0f};
            v2f B = {bx, by};
            acc2 = __builtin_amdgcn_wmma_f32_16x16x4_f32(false, A, false, B,
                                                         (short)0, acc2, false, false);
        }
        {
            float bx = Zin[(int)(((unsigned)p3.x) >> 5) * 16 + mn];
            float by = Zin[(int)(((unsigned)p3.y) >> 5) * 16 + mn];
            v2f A = {((p3.x & 31) == mn) ? 1.0f : 0.0f,
                     ((p3.y & 31) == mn) ? 1.0f : 0.0f};
            v2f B = {bx, by};
            acc3 = __builtin_amdgcn_wmma_f32_16x16x4_f32(false, A, false, B,
                                                         (short)0, acc3, false, false);
        }
    }

    v8f acc = (acc0 + acc1) + (acc2 + acc3);

    // Fused combine: Y' = clip(C1*Y + C2*agg*dinv[dst] + C3*X), mask-freeze,
    // and Z' = Y' * dinv for the next step.
    // C layout: vgpr j, lanes 0-15 -> node row j ; lanes 16-31 -> node row j+8.
    const int nbase = (b << 4) + (half << 3);
    if (((b << 4) | 15) < N) {
        // full bucket: no per-row bounds checks
        #pragma unroll
        for (int j = 0; j < 8; ++j) {
            int node = nbase + j;
            float di = dinv[node];
            int o = node * 16 + mn;
            float xv = X[o];
            float y = C1 * Yin[o] + C2 * (acc[j] * di) + C3 * xv;
            y = fminf(1.0f, fmaxf(-1.0f, y));
            if (mask[node]) y = xv;
            Yout[o] = y;
            Zout[o] = y * di;
        }
    } else {
        #pragma unroll
        for (int j = 0; j < 8; ++j) {
            int node = nbase + j;
            if (node < N) {
                float di = dinv[node];
                int o = node * 16 + mn;
                float xv = X[o];
                float y = C1 * Yin[o] + C2 * (acc[j] * di) + C3 * xv;
                y = fminf(1.0f, fmaxf(-1.0f, y));
                if (mask[node]) y = xv;
                Yout[o] = y;
                Zout[o] = y * di;
            }
        }
    }
}

// ---------------- host launcher ----------------

static inline size_t alignup256(size_t x) { return (x + 255) & ~(size_t)255; }

extern "C" void kernel_launch(void* const* d_in, const int* in_sizes, int n_in,
                              void* d_out, int out_size, void* d_ws, size_t ws_size,
                              hipStream_t stream) {
    (void)n_in; (void)out_size; (void)ws_size;

    const int E = in_sizes[0] / 2;
    const int N = in_sizes[1] / 16;
    const int NB = (N + 15) >> 4;
    const int EP = E + BPAD * NB + 64;  // padded edge buffer capacity (+prefetch slack)

    const int* edge = (const int*)d_in[0];
    const int* src  = edge;
    const int* dst  = edge + E;
    const float* X  = (const float*)d_in[1];
    const unsigned char* mask = (const unsigned char*)d_in[2];
    float* out = (float*)d_out;

    // carve workspace
    char* w = (char*)d_ws;
    int*   deg    = (int*)w;   w += alignup256((size_t)N * 4);
    float* dinv   = (float*)w; w += alignup256((size_t)N * 4);
    int*   boff   = (int*)w;   w += alignup256((size_t)(NB + 1) * 4);
    int*   cursor = (int*)w;   w += alignup256((size_t)NB * 4);
    int*   bsz    = (int*)w;   w += alignup256((size_t)NB * 4);
    int*   ebuf   = (int*)w;   w += alignup256((size_t)EP * 4);
    float* Ya     = (float*)w; w += alignup256((size_t)N * 16 * 4);
    float* Yb     = (float*)w; w += alignup256((size_t)N * 16 * 4);
    float* Za     = (float*)w; w += alignup256((size_t)N * 16 * 4);
    float* Zb     = (float*)w; w += alignup256((size_t)N * 16 * 4);

    const int TB = 256;
    hipMemsetAsync(deg, 0, (size_t)N * 4, stream);
    deg_kernel<<<(E + TB - 1) / TB, TB, 0, stream>>>(dst, deg, E);
    scan_kernel<<<1, 1024, 0, stream>>>(deg, boff, cursor, bsz, NB, N);
    fill_kernel<<<(E + TB - 1) / TB, TB, 0, stream>>>(src, dst, cursor, ebuf, E);
    pad_kernel<<<(16 * NB + TB - 1) / TB, TB, 0, stream>>>(boff, bsz, ebuf, NB);
    init_kernel<<<(N + TB - 1) / TB, TB, 0, stream>>>(X, deg, dinv, Ya, Za, N);

    const int WPB = TB / 32;                 // 8 waves (buckets) per block
    const int pgrid = (NB + WPB - 1) / WPB;

    float* Yi = Ya; float* Zi = Za;
    float* Yo = Yb; float* Zo = Zb;
    for (int s = 0; s < PROP_STEPS; ++s) {
        float* Ywrite = (s == PROP_STEPS - 1) ? out : Yo;
        prop_kernel<<<pgrid, TB, 0, stream>>>(boff, ebuf, Zi, Yi, X, mask, dinv,
                                              Ywrite, Zo, NB, N);
        float* t;
        t = Yi; Yi = Ywrite; Yo = t;
        t = Zi; Zi = Zo;     Zo = t;
    }
}